// GCNLayer_3779571220516
// MI455X (gfx1250) — compile-verified
//
#include <hip/hip_runtime.h>

typedef __attribute__((ext_vector_type(2))) float v2f;
typedef __attribute__((ext_vector_type(8))) float v8f;

#define NVERT 50000
#define KNBR  16
#define DIM   256
#define HID   256
#define MBLK  64            // vertices per workgroup
#define NTHREADS 512        // 16 waves (wave32)
#define LDS_STRIDE 260      // 256 + 4 pad -> conflict-free ds_load_b64 A fragments

__global__ __launch_bounds__(NTHREADS)
void gcn_fused_wmma_kernel(const float* __restrict__ vf,
                           const int*   __restrict__ nidx,
                           const int*   __restrict__ vlen,
                           const float* __restrict__ Wg,
                           const float* __restrict__ Bg,
                           float*       __restrict__ out)
{
    // 2 * 64 * 260 * 4 = 133,120 bytes static LDS (gfx1250 WGP has 320 KB)
    __shared__ float ldsAGG[MBLK * LDS_STRIDE];
    __shared__ float ldsVF [MBLK * LDS_STRIDE];

    const int tid   = threadIdx.x;
    const int vbase = blockIdx.x * MBLK;

    // ---------------- Phase 1: masked-mean aggregation + vf staging ----------------
    {
        const int  i  = tid >> 3;      // vertex within tile: 0..63
        const int  s  = tid & 7;       // 32-float slice:      0..7
        const int  v  = vbase + i;
        const bool ok = (v < NVERT);
        int vl = ok ? vlen[v] : 0;
        if (vl > KNBR) vl = KNBR;
        if (vl < 0)    vl = 0;

        // stage this vertex's own features into LDS
        const float4* vrow = (const float4*)(vf + (size_t)(ok ? v : 0) * DIM + s * 32);
        float* dstv = ldsVF + i * LDS_STRIDE + s * 32;
        #pragma unroll
        for (int j = 0; j < 8; ++j) {
            float4 t = ok ? vrow[j] : make_float4(0.f, 0.f, 0.f, 0.f);
            *(float4*)(dstv + 4 * j) = t;
        }

        float4 acc[8];
        #pragma unroll
        for (int j = 0; j < 8; ++j) acc[j] = make_float4(0.f, 0.f, 0.f, 0.f);

        const int* nb = nidx + (size_t)(ok ? v : 0) * KNBR;
        for (int k = 0; k < vl; ++k) {
            const int u = nb[k];
            const float4* nrow = (const float4*)(vf + (size_t)u * DIM + s * 32);
            #pragma unroll
            for (int j = 0; j < 8; ++j) {
                float4 t = nrow[j];
                acc[j].x += t.x; acc[j].y += t.y; acc[j].z += t.z; acc[j].w += t.w;
            }
        }

        const float scale = 1.0f / (float)(vl > 0 ? vl : 1);
        float* dsta = ldsAGG + i * LDS_STRIDE + s * 32;
        #pragma unroll
        for (int j = 0; j < 8; ++j) {
            float4 t = acc[j];
            t.x *= scale; t.y *= scale; t.z *= scale; t.w *= scale;
            *(float4*)(dsta + 4 * j) = t;
        }
    }
    __syncthreads();

    // ---------------- Phase 2: fused dual GEMM, f32 WMMA 16x16x4 ----------------
    // wave -> 16 output columns; 4 M-subtiles per wave (B fragments reused x4)
    const int wave  = tid >> 5;      // 0..15
    const int lane  = tid & 31;
    const int hi    = lane >> 4;     // K half selector (A/B fragment layout, ISA 7.12.2)
    const int ln    = lane & 15;     // M (A) / N (B,C) index within fragment
    const int nbase = wave * 16;

    v8f c0 = {}, c1 = {}, c2 = {}, c3 = {};

    const float* wcol = Wg + nbase + ln;
    const float* bcol = Bg + nbase + ln;
    const float* arow0 = ldsAGG + (0 * 16 + ln) * LDS_STRIDE;
    const float* arow1 = ldsAGG + (1 * 16 + ln) * LDS_STRIDE;
    const float* arow2 = ldsAGG + (2 * 16 + ln) * LDS_STRIDE;
    const float* arow3 = ldsAGG + (3 * 16 + ln) * LDS_STRIDE;
    const float* frow0 = ldsVF  + (0 * 16 + ln) * LDS_STRIDE;
    const float* frow1 = ldsVF  + (1 * 16 + ln) * LDS_STRIDE;
    const float* frow2 = ldsVF  + (2 * 16 + ln) * LDS_STRIDE;
    const float* frow3 = ldsVF  + (3 * 16 + ln) * LDS_STRIDE;

    #pragma unroll 2
    for (int k0 = 0; k0 < DIM; k0 += 4) {
        const int kk = k0 + 2 * hi;

        v2f bw, bb;
        bw.x = wcol[(size_t)kk * HID];
        bw.y = wcol[(size_t)(kk + 1) * HID];
        bb.x = bcol[(size_t)kk * HID];
        bb.y = bcol[(size_t)(kk + 1) * HID];

        v2f a0 = *(const v2f*)(arow0 + kk);
        v2f a1 = *(const v2f*)(arow1 + kk);
        v2f a2 = *(const v2f*)(arow2 + kk);
        v2f a3 = *(const v2f*)(arow3 + kk);
        v2f f0 = *(const v2f*)(frow0 + kk);
        v2f f1 = *(const v2f*)(frow1 + kk);
        v2f f2 = *(const v2f*)(frow2 + kk);
        v2f f3 = *(const v2f*)(frow3 + kk);

        c0 = __builtin_amdgcn_wmma_f32_16x16x4_f32(false, a0, false, bw, (short)0, c0, false, false);
        c1 = __builtin_amdgcn_wmma_f32_16x16x4_f32(false, a1, false, bw, (short)0, c1, false, false);
        c2 = __builtin_amdgcn_wmma_f32_16x16x4_f32(false, a2, false, bw, (short)0, c2, false, false);
        c3 = __builtin_amdgcn_wmma_f32_16x16x4_f32(false, a3, false, bw, (short)0, c3, false, false);
        c0 = __builtin_amdgcn_wmma_f32_16x16x4_f32(false, f0, false, bb, (short)0, c0, false, false);
        c1 = __builtin_amdgcn_wmma_f32_16x16x4_f32(false, f1, false, bb, (short)0, c1, false, false);
        c2 = __builtin_amdgcn_wmma_f32_16x16x4_f32(false, f2, false, bb, (short)0, c2, false, false);
        c3 = __builtin_amdgcn_wmma_f32_16x16x4_f32(false, f3, false, bb, (short)0, c3, false, false);
    }

    // ---------------- Epilogue: ReLU + store ----------------
    // C/D layout: VGPR j, lanes 0-15 -> row j, lanes 16-31 -> row j+8; col = ln
    const int col = nbase + ln;
    #pragma unroll
    for (int mi = 0; mi < 4; ++mi) {
        const v8f c = (mi == 0) ? c0 : (mi == 1) ? c1 : (mi == 2) ? c2 : c3;
        #pragma unroll
        for (int j = 0; j < 8; ++j) {
            const int m = mi * 16 + j + 8 * hi;
            const int v = vbase + m;
            if (v < NVERT) {
                out[(size_t)v * HID + col] = __builtin_fmaxf(c[j], 0.0f);
            }
        }
    }
}

extern "C" void kernel_launch(void* const* d_in, const int* in_sizes, int n_in,
                              void* d_out, int out_size, void* d_ws, size_t ws_size,
                              hipStream_t stream) {
    (void)in_sizes; (void)n_in; (void)d_ws; (void)ws_size; (void)out_size;
    const float* vf   = (const float*)d_in[0];   // [N, D]
    const int*   nidx = (const int*)  d_in[1];   // [N, K]
    const int*   vlen = (const int*)  d_in[2];   // [N]
    const float* Wg   = (const float*)d_in[3];   // [D, H]
    const float* Bg   = (const float*)d_in[4];   // [D, H]
    float*       out  = (float*)d_out;           // [N, H]

    const int grid = (NVERT + MBLK - 1) / MBLK;  // 782
    gcn_fused_wmma_kernel<<<grid, NTHREADS, 0, stream>>>(vf, nidx, vlen, Wg, Bg, out);
}